// AggregationLayer_59502476919113
// MI455X (gfx1250) — compile-verified
//
#include <hip/hip_runtime.h>

// CDNA5 segment-mean via WMMA "one-hot matmul" scatter-add.
// D = A(one-hot 16x4) x B(gathered rows 4x16) + C, accumulated over the
// sorted edge range owned by each wave's 16 segments. No atomics.
// Loop bounds are readfirstlane'd (wave-uniform) so the edge loop is a scalar
// branch and EXEC stays all-1s across every v_wmma (ISA requirement).

typedef __attribute__((ext_vector_type(2))) float v2f;
typedef __attribute__((ext_vector_type(8))) float v8f;

#define N_SRC   200000
#define N_EDGES 1280000
#define N_SEG   100000
#define DFEAT   64
#define SEG_PER_WAVE 16
#define WAVES_PER_BLOCK 8

__global__ __launch_bounds__(256) void seg_mean_wmma_kernel(
    const float* __restrict__ values,       // [N_SRC, 64] f32
    const int*   __restrict__ gather_idx,   // [N_EDGES] i32
    const int*   __restrict__ segment_ids,  // [N_EDGES] i32, sorted ascending
    float*       __restrict__ out)          // [N_SEG, 64] f32
{
    const int lane = threadIdx.x & 31;
    const int wave = blockIdx.x * WAVES_PER_BLOCK + (threadIdx.x >> 5);

    // Wave-uniform segment base (readfirstlane => compiler treats as scalar).
    const int S = __builtin_amdgcn_readfirstlane(wave * SEG_PER_WAVE);
    if (S >= N_SEG) return;                 // uniform: whole wave exits together

    // ---- scalar binary searches: edge range [e_lo, e_hi) for segs [S, S+16) ----
    int lo = 0, hi = N_EDGES;
    while (lo < hi) { int m = (lo + hi) >> 1; if (segment_ids[m] < S) lo = m + 1; else hi = m; }
    const int e_lo = __builtin_amdgcn_readfirstlane(lo);
    lo = e_lo; hi = N_EDGES;
    const int Send = S + SEG_PER_WAVE;
    while (lo < hi) { int m = (lo + hi) >> 1; if (segment_ids[m] < Send) lo = m + 1; else hi = m; }
    const int e_hi = __builtin_amdgcn_readfirstlane(lo);

    // ---- lane roles per ISA 16x16x4 f32 layouts ----
    const int half = lane >> 4;        // 0: K={0,1}, 1: K={2,3}
    const int col  = lane & 15;        // A row (segment) / B column (feature)
    const int k0   = half * 2;
    const int k1   = half * 2 + 1;

    v8f acc0 = {}, acc1 = {}, acc2 = {}, acc3 = {};   // 16 segs x 64 feats
    float cntp = 0.f;                                 // per-segment edge count partial

    const int last = e_hi - 1;                        // valid whenever loop runs
    for (int e = e_lo; e < e_hi; e += 4) {            // scalar loop (uniform bounds)
        const int i0 = e + k0, i1 = e + k1;
        const bool ok0 = i0 < e_hi, ok1 = i1 < e_hi;
        const int c0 = ok0 ? i0 : last;               // clamp: branch-free tail
        const int c1 = ok1 ? i1 : last;

        const int s0 = segment_ids[c0];               // unconditional loads
        const int s1 = segment_ids[c1];

        v2f a;                                        // one-hot, masked by validity
        a.x = (ok0 && s0 == S + col) ? 1.0f : 0.0f;
        a.y = (ok1 && s1 == S + col) ? 1.0f : 0.0f;
        cntp += a.x + a.y;

        // 32-bit offsets: values table is 12.8M floats, fits u32 => saddr+voffset
        const unsigned o0 = ((unsigned)gather_idx[c0] << 6) + (unsigned)col;
        const unsigned o1 = ((unsigned)gather_idx[c1] << 6) + (unsigned)col;

        v2f b0, b1, b2, b3;                           // 4 feature tiles of 16 cols
        b0.x = values[o0 +  0]; b0.y = values[o1 +  0];
        b1.x = values[o0 + 16]; b1.y = values[o1 + 16];
        b2.x = values[o0 + 32]; b2.y = values[o1 + 32];
        b3.x = values[o0 + 48]; b3.y = values[o1 + 48];

        acc0 = __builtin_amdgcn_wmma_f32_16x16x4_f32(false, a, false, b0, (short)0, acc0, false, false);
        acc1 = __builtin_amdgcn_wmma_f32_16x16x4_f32(false, a, false, b1, (short)0, acc1, false, false);
        acc2 = __builtin_amdgcn_wmma_f32_16x16x4_f32(false, a, false, b2, (short)0, acc2, false, false);
        acc3 = __builtin_amdgcn_wmma_f32_16x16x4_f32(false, a, false, b3, (short)0, acc3, false, false);
    }

    // ---- counts: lane L and L^16 both accumulated segment (S + col) ----
    float cnt = cntp + __shfl_xor(cntp, 16, 32);      // all lanes: count of seg S+col

    // ---- mean + store: C/D VGPR j holds seg row j (lanes 0-15) / j+8 (16-31) ----
#pragma unroll
    for (int j = 0; j < 8; ++j) {
        const int m = half ? (j + 8) : j;             // segment row of this slice
        const float c = __shfl(cnt, m, 32);           // count lives in lane m (m<16)
        const float r = 1.0f / fmaxf(c, 1.0f);
        float* o = out + (long)(S + m) * DFEAT + col;
        o[ 0] = acc0[j] * r;
        o[16] = acc1[j] * r;
        o[32] = acc2[j] * r;
        o[48] = acc3[j] * r;
    }
}

extern "C" void kernel_launch(void* const* d_in, const int* in_sizes, int n_in,
                              void* d_out, int out_size, void* d_ws, size_t ws_size,
                              hipStream_t stream) {
    (void)in_sizes; (void)n_in; (void)d_ws; (void)ws_size; (void)out_size;
    const float* values      = (const float*)d_in[0];
    const int*   gather_idx  = (const int*)d_in[1];
    const int*   segment_ids = (const int*)d_in[2];
    float*       out         = (float*)d_out;

    const int wave_groups = (N_SEG + SEG_PER_WAVE - 1) / SEG_PER_WAVE;        // 6250 waves
    const int blocks = (wave_groups + WAVES_PER_BLOCK - 1) / WAVES_PER_BLOCK; // 782 blocks
    hipLaunchKernelGGL(seg_mean_wmma_kernel, dim3(blocks), dim3(256), 0, stream,
                       values, gather_idx, segment_ids, out);
}